// GATNet_36627481101157
// MI455X (gfx1250) — compile-verified
//
#include <hip/hip_runtime.h>
#include <math.h>

// GAT 2-layer forward for MI455X (gfx1250, wave32).
// GEMMs use V_WMMA_F32_16X16X4_F32 (fp32 matrix pipe) since the workload is
// dominated by the edge gather/scatter phase, not the GEMMs -> keep f32.

typedef __attribute__((ext_vector_type(2))) float v2f;
typedef __attribute__((ext_vector_type(8))) float v8f;

#define H1C 8
#define D1C 8
#define F1  128   // in features
#define FM  64    // H1*D1 (mid features)
#define C2  40    // out classes
#define C2P 48    // padded to 3x16 WMMA tiles

// ---- float atomic max via int/uint ordering trick (valid for mixed signs,
// init value -inf works in both orderings) ----
__device__ __forceinline__ void atomic_max_f32(float* addr, float v) {
  if (v >= 0.0f) atomicMax((int*)addr, __float_as_int(v));
  else           atomicMin((unsigned int*)addr, __float_as_uint(v));
}

__device__ __forceinline__ void edge_sd(const int* __restrict__ ei, int e, int E,
                                        int& s, int& d) {
  if (e < E) { s = ei[e]; d = ei[E + e]; }
  else       { s = e - E; d = s; }        // appended self-loops
}

__global__ void k_fill(float* p, float v, int n) {
  int i = blockIdx.x * blockDim.x + threadIdx.x;
  if (i < n) p[i] = v;
}

// ---- Layer 1 GEMM: h1[N,64] = x[N,128] @ W1[128,64] ------------------------
// 128 threads = 4 waves; each wave owns a 16x16 tile of the 16-row block.
__global__ void k_gemm1(const float* __restrict__ x, const float* __restrict__ W,
                        float* __restrict__ h1) {
  __shared__ float lw[F1 * FM];                 // 32 KB
  int tid = threadIdx.x;
  for (int i = tid; i < F1 * FM; i += 128) lw[i] = W[i];
  __syncthreads();
  int wave = tid >> 5, lane = tid & 31;
  int hi = lane >> 4, ln = lane & 15;
  int row0 = blockIdx.x * 16;
  int col0 = wave * 16;
  const float* xr = x + (size_t)(row0 + ln) * F1;
  v8f c = {};
  for (int k = 0; k < F1; k += 4) {
    // A 16x4: lanes0-15 K={0,1}, lanes16-31 K={2,3}; row = lane&15
    v2f a = *(const v2f*)(xr + k + 2 * hi);
    v2f b;
    b.x = lw[(k + 2 * hi) * FM + col0 + ln];
    b.y = lw[(k + 2 * hi + 1) * FM + col0 + ln];
    c = __builtin_amdgcn_wmma_f32_16x16x4_f32(false, a, false, b,
                                              (short)0, c, false, false);
  }
  // C/D: VGPR j -> M = j + 8*hi, N = lane&15
  for (int j = 0; j < 8; ++j)
    h1[(size_t)(row0 + j + 8 * hi) * FM + col0 + ln] = c[j];
}

// ---- Layer 2 GEMM: h2[N,40] = hmid[N,64] @ W2[64,40] (padded to 48 cols) ---
__global__ void k_gemm2(const float* __restrict__ h, const float* __restrict__ W2,
                        float* __restrict__ h2) {
  __shared__ float lw[FM * C2P];                // 12 KB, zero padded
  int tid = threadIdx.x;                        // 96 threads = 3 waves
  for (int i = tid; i < FM * C2P; i += 96) {
    int k = i / C2P, cc = i - k * C2P;
    lw[i] = (cc < C2) ? W2[k * C2 + cc] : 0.0f;
  }
  __syncthreads();
  int wave = tid >> 5, lane = tid & 31;
  int hi = lane >> 4, ln = lane & 15;
  int row0 = blockIdx.x * 16;
  int col0 = wave * 16;
  const float* hr = h + (size_t)(row0 + ln) * FM;
  v8f c = {};
  for (int k = 0; k < FM; k += 4) {
    v2f a = *(const v2f*)(hr + k + 2 * hi);
    v2f b;
    b.x = lw[(k + 2 * hi) * C2P + col0 + ln];
    b.y = lw[(k + 2 * hi + 1) * C2P + col0 + ln];
    c = __builtin_amdgcn_wmma_f32_16x16x4_f32(false, a, false, b,
                                              (short)0, c, false, false);
  }
  int col = col0 + ln;
  if (col < C2)
    for (int j = 0; j < 8; ++j)
      h2[(size_t)(row0 + j + 8 * hi) * C2 + col] = c[j];
}

// ---- per-node attention logits ---------------------------------------------
__global__ void k_attn1(const float* __restrict__ h1, const float* __restrict__ as,
                        const float* __restrict__ ad, float* es, float* ed, int n) {
  int i = blockIdx.x * blockDim.x + threadIdx.x;   // n*8
  if (i >= n * H1C) return;
  int node = i >> 3, hh = i & 7;
  const float* hp = h1 + (size_t)node * FM + hh * D1C;
  float s = 0.f, d = 0.f;
  for (int j = 0; j < D1C; ++j) {
    s += hp[j] * as[hh * D1C + j];
    d += hp[j] * ad[hh * D1C + j];
  }
  es[i] = s; ed[i] = d;
}

__global__ void k_attn2(const float* __restrict__ h2, const float* __restrict__ as,
                        const float* __restrict__ ad, float* es, float* ed, int n) {
  int i = blockIdx.x * blockDim.x + threadIdx.x;
  if (i >= n) return;
  const float* hp = h2 + (size_t)i * C2;
  float s = 0.f, d = 0.f;
  for (int j = 0; j < C2; ++j) { s += hp[j] * as[j]; d += hp[j] * ad[j]; }
  es[i] = s; ed[i] = d;
}

// ---- layer-1 edge passes (thread per (edge, head)) -------------------------
__global__ void k_emax1(const int* __restrict__ ei, const float* __restrict__ es,
                        const float* __restrict__ ed, float* m, int E, int ET) {
  int i = blockIdx.x * blockDim.x + threadIdx.x;
  if (i >= ET * H1C) return;
  int e = i >> 3, hh = i & 7;
  int s, d; edge_sd(ei, e, E, s, d);
  float t = es[s * H1C + hh] + ed[d * H1C + hh];
  t = t > 0.f ? t : 0.2f * t;
  atomic_max_f32(&m[d * H1C + hh], t);
}

__global__ void k_eden1(const int* __restrict__ ei, const float* __restrict__ es,
                        const float* __restrict__ ed, const float* __restrict__ m,
                        float* den, int E, int ET) {
  int i = blockIdx.x * blockDim.x + threadIdx.x;
  if (i >= ET * H1C) return;
  int e = i >> 3, hh = i & 7;
  int s, d; edge_sd(ei, e, E, s, d);
  int d8 = d * H1C + hh;
  float t = es[s * H1C + hh] + ed[d8];
  t = t > 0.f ? t : 0.2f * t;
  atomicAdd(&den[d8], expf(t - m[d8]));
}

__global__ void k_eagg1(const int* __restrict__ ei, const float* __restrict__ es,
                        const float* __restrict__ ed, const float* __restrict__ m,
                        const float* __restrict__ den, const float* __restrict__ h1,
                        float* agg, int E, int ET) {
  int i = blockIdx.x * blockDim.x + threadIdx.x;
  if (i >= ET * H1C) return;
  int e = i >> 3, hh = i & 7;
  int s, d; edge_sd(ei, e, E, s, d);
  int d8 = d * H1C + hh;
  float t = es[s * H1C + hh] + ed[d8];
  t = t > 0.f ? t : 0.2f * t;
  float alpha = expf(t - m[d8]) / den[d8];
  const float* hp = h1 + (size_t)s * FM + hh * D1C;
  float* op = agg + (size_t)d * FM + hh * D1C;
  for (int j = 0; j < D1C; ++j) atomicAdd(&op[j], alpha * hp[j]);
}

__global__ void k_elu(const float* __restrict__ agg, const float* __restrict__ b,
                      float* out, int n) {
  int i = blockIdx.x * blockDim.x + threadIdx.x;
  if (i >= n * FM) return;
  float v = agg[i] + b[i & (FM - 1)];
  out[i] = v > 0.f ? v : (expf(v) - 1.0f);
}

// ---- layer-2 edge passes (1 head) ------------------------------------------
__global__ void k_emax2(const int* __restrict__ ei, const float* __restrict__ es,
                        const float* __restrict__ ed, float* m, int E, int ET) {
  int i = blockIdx.x * blockDim.x + threadIdx.x;
  if (i >= ET) return;
  int s, d; edge_sd(ei, i, E, s, d);
  float t = es[s] + ed[d];
  t = t > 0.f ? t : 0.2f * t;
  atomic_max_f32(&m[d], t);
}

__global__ void k_eden2(const int* __restrict__ ei, const float* __restrict__ es,
                        const float* __restrict__ ed, const float* __restrict__ m,
                        float* den, int E, int ET) {
  int i = blockIdx.x * blockDim.x + threadIdx.x;
  if (i >= ET) return;
  int s, d; edge_sd(ei, i, E, s, d);
  float t = es[s] + ed[d];
  t = t > 0.f ? t : 0.2f * t;
  atomicAdd(&den[d], expf(t - m[d]));
}

__global__ void k_eagg2(const int* __restrict__ ei, const float* __restrict__ es,
                        const float* __restrict__ ed, const float* __restrict__ m,
                        const float* __restrict__ den, const float* __restrict__ h2,
                        float* agg, int E, int ET) {
  int i = blockIdx.x * blockDim.x + threadIdx.x;  // ET * 5 (8 channels each)
  if (i >= ET * 5) return;
  int e = i / 5, cb = i - e * 5;
  int s, d; edge_sd(ei, e, E, s, d);
  float t = es[s] + ed[d];
  t = t > 0.f ? t : 0.2f * t;
  float alpha = expf(t - m[d]) / den[d];
  int c0 = cb * 8;
  const float* hp = h2 + (size_t)s * C2 + c0;
  float* op = agg + (size_t)d * C2 + c0;
  for (int j = 0; j < 8; ++j) atomicAdd(&op[j], alpha * hp[j]);
}

// ---- bias + log_softmax -----------------------------------------------------
__global__ void k_final(const float* __restrict__ agg, const float* __restrict__ b,
                        float* __restrict__ out, int n) {
  int i = blockIdx.x * blockDim.x + threadIdx.x;
  if (i >= n) return;
  float v[C2];
  float mx = -INFINITY;
  for (int c = 0; c < C2; ++c) {
    v[c] = agg[(size_t)i * C2 + c] + b[c];
    mx = fmaxf(mx, v[c]);
  }
  float sum = 0.f;
  for (int c = 0; c < C2; ++c) sum += expf(v[c] - mx);
  float ls = logf(sum);
  for (int c = 0; c < C2; ++c) out[(size_t)i * C2 + c] = v[c] - mx - ls;
}

static inline unsigned gdiv(long long n, int b) { return (unsigned)((n + b - 1) / b); }

extern "C" void kernel_launch(void* const* d_in, const int* in_sizes, int n_in,
                              void* d_out, int out_size, void* d_ws, size_t ws_size,
                              hipStream_t stream) {
  const float* x   = (const float*)d_in[0];
  const int*   ei  = (const int*)  d_in[1];
  const float* W1  = (const float*)d_in[2];
  const float* as1 = (const float*)d_in[3];
  const float* ad1 = (const float*)d_in[4];
  const float* b1  = (const float*)d_in[5];
  const float* W2  = (const float*)d_in[6];
  const float* as2 = (const float*)d_in[7];
  const float* ad2 = (const float*)d_in[8];
  const float* b2  = (const float*)d_in[9];
  float* out = (float*)d_out;

  const int N  = in_sizes[0] / F1;   // 50000 (divisible by 16)
  const int E  = in_sizes[1] / 2;    // 1600000
  const int ET = E + N;              // + self loops

  // workspace layout (floats): ~308*N = ~62 MB
  float* ws   = (float*)d_ws;
  float* h1   = ws;                            // N*64
  float* agg1 = h1   + (size_t)N * FM;         // N*64
  float* hmid = agg1 + (size_t)N * FM;         // N*64
  float* h2   = hmid + (size_t)N * FM;         // N*40
  float* agg2 = h2   + (size_t)N * C2;         // N*40
  float* es1  = agg2 + (size_t)N * C2;         // N*8
  float* ed1  = es1  + (size_t)N * H1C;
  float* m1   = ed1  + (size_t)N * H1C;
  float* den1 = m1   + (size_t)N * H1C;
  float* es2  = den1 + (size_t)N * H1C;        // N
  float* ed2  = es2  + N;
  float* m2   = ed2  + N;
  float* den2 = m2   + N;

  const int B = 256;
  const float NEG_INF = -INFINITY;

  // init (must run every launch: graph replay + poisoned ws)
  k_fill<<<gdiv((long long)N * H1C, B), B, 0, stream>>>(m1, NEG_INF, N * H1C);
  k_fill<<<gdiv((long long)N * H1C, B), B, 0, stream>>>(den1, 0.f, N * H1C);
  k_fill<<<gdiv((long long)N * FM, B), B, 0, stream>>>(agg1, 0.f, N * FM);
  k_fill<<<gdiv(N, B), B, 0, stream>>>(m2, NEG_INF, N);
  k_fill<<<gdiv(N, B), B, 0, stream>>>(den2, 0.f, N);
  k_fill<<<gdiv((long long)N * C2, B), B, 0, stream>>>(agg2, 0.f, N * C2);

  // layer 1
  k_gemm1<<<N / 16, 128, 0, stream>>>(x, W1, h1);
  k_attn1<<<gdiv((long long)N * H1C, B), B, 0, stream>>>(h1, as1, ad1, es1, ed1, N);
  k_emax1<<<gdiv((long long)ET * H1C, B), B, 0, stream>>>(ei, es1, ed1, m1, E, ET);
  k_eden1<<<gdiv((long long)ET * H1C, B), B, 0, stream>>>(ei, es1, ed1, m1, den1, E, ET);
  k_eagg1<<<gdiv((long long)ET * H1C, B), B, 0, stream>>>(ei, es1, ed1, m1, den1, h1, agg1, E, ET);
  k_elu<<<gdiv((long long)N * FM, B), B, 0, stream>>>(agg1, b1, hmid, N);

  // layer 2
  k_gemm2<<<N / 16, 96, 0, stream>>>(hmid, W2, h2);
  k_attn2<<<gdiv(N, B), B, 0, stream>>>(h2, as2, ad2, es2, ed2, N);
  k_emax2<<<gdiv(ET, B), B, 0, stream>>>(ei, es2, ed2, m2, E, ET);
  k_eden2<<<gdiv(ET, B), B, 0, stream>>>(ei, es2, ed2, m2, den2, E, ET);
  k_eagg2<<<gdiv((long long)ET * 5, B), B, 0, stream>>>(ei, es2, ed2, m2, den2, h2, agg2, E, ET);

  // output
  k_final<<<gdiv(N, B), B, 0, stream>>>(agg2, b2, out, N);
}